// SDFGridLoss_12524124635645
// MI455X (gfx1250) — compile-verified
//
#include <hip/hip_runtime.h>

typedef float v2f __attribute__((ext_vector_type(2)));
typedef float v8f __attribute__((ext_vector_type(8)));

// global-address-space pointer (forces global_load_* instead of flat_load_*)
typedef const float __attribute__((address_space(1)))* gcfloat_p;

#define B_   4
#define S_   4096
#define N_   4096
#define WAVES 4
#define TPB  (WAVES * 32)
#define TILES (S_ / 16)

__device__ __forceinline__ gcfloat_p as_global(const float* p) {
    // inttoptr round-trip: no addrspacecast needed, yields AS(1) loads
    return (gcfloat_p)(uintptr_t)p;
}

// raw v_min_num_f32: skips the sNaN-canonicalize (v_max_num x,x,x) fminf emits
__device__ __forceinline__ float fmin_raw(float a, float b) {
    float r;
    asm("v_min_num_f32 %0, %1, %2" : "=v"(r) : "v"(a), "v"(b));
    return r;
}

// One workgroup = one (batch, 16-grid-point tile). Each wave scans a slab of
// the cloud 16 points at a time with V_WMMA_F32_16X16X4_F32. The unused 4th
// K-slot injects |y|^2:  A[M][:] = (-2x,-2y,-2z, 1), B[:][N] = (yx,yy,yz,|y|^2)
//   => D[M][N] = |y_N|^2 - 2 x_M . y_N  (= d^2 - |x_M|^2), C = inline 0.
__global__ __launch_bounds__(TPB) void sdf_min_kernel(
    const float* __restrict__ gp,     // [B,S,3]
    const float* __restrict__ gts,    // [B,N,3]
    const float* __restrict__ preds,  // [B,N,3]
    float* __restrict__ ws)           // [B, TILES] partial sums
{
    const int lane    = threadIdx.x & 31;
    // wave index is wave-uniform: pin it to an SGPR so loop control stays SALU
    const int wave    = __builtin_amdgcn_readfirstlane(threadIdx.x >> 5);
    const int sTile   = blockIdx.x;   // 0..TILES-1
    const int b       = blockIdx.y;   // 0..B-1
    const int s0      = sTile * 16;
    const int laneMod = lane & 15;
    const bool loHalf = lane < 16;

    // ---- load this lane's grid point (row M = laneMod of the tile) ----
    gcfloat_p g = as_global(gp) + ((size_t)b * S_ + (s0 + laneMod)) * 3;
    const float gx = g[0], gy = g[1], gz = g[2];

    // A fragment 16x4 (M x K): K = {-2x, -2y, -2z, 1}
    v2f afrag;
    if (loHalf) { afrag[0] = -2.0f * gx; afrag[1] = -2.0f * gy; }
    else        { afrag[0] = -2.0f * gz; afrag[1] = 1.0f;       }
    // partial |x|^2: halves hold (gx^2+gy^2) and (gz^2) for the same point
    const float p = loHalf ? (gx * gx + gy * gy) : (gz * gz);

    v8f czero = {};  // SRC2 = inline 0

    __shared__ float lmin[2][WAVES][16];

    const float* clouds[2] = { preds, gts };
    const int nPerWave = N_ / WAVES;
    const int nIters   = nPerWave / 16;
    const int nBase    = wave * nPerWave;

    #pragma unroll
    for (int c = 0; c < 2; ++c) {
        // uniform SGPR base + 32-bit per-lane element offset  -> SADDR+VOFFSET
        gcfloat_p clBase = as_global(clouds[c]) + (size_t)b * N_ * 3;
        unsigned off = (unsigned)((nBase + laneMod) * 3);

        float md[8];
        #pragma unroll
        for (int i = 0; i < 8; ++i) md[i] = 3.0e38f;

        #pragma unroll 4
        for (int it = 0; it < nIters; ++it, off += 48) {
            const float cx = clBase[off + 0];
            const float cy = clBase[off + 1];
            const float cz = clBase[off + 2];
            float y2 = cx * cx + cy * cy + cz * cz;
            // tie y2 to a VGPR in ALL lanes so the select below stays a
            // v_cndmask instead of an exec-mask branch around its computation
            asm("" : "+v"(y2));

            // B fragment 4x16 (K x N), column N = laneMod
            v2f bfrag;
            bfrag[0] = loHalf ? cx : cz;
            bfrag[1] = loHalf ? cy : y2;

            // D = A x B + 0  ->  |y|^2 - 2 x.y  for a 16x16 tile
            v8f d = __builtin_amdgcn_wmma_f32_16x16x4_f32(
                false, afrag, false, bfrag, (short)0, czero, false, false);

            #pragma unroll
            for (int i = 0; i < 8; ++i) md[i] = fmin_raw(md[i], d[i]);
        }

        // add |x_M|^2 (constant over the min): row M = i (lanes<16) or 8+i
        #pragma unroll
        for (int i = 0; i < 8; ++i) {
            const int m = loHalf ? i : (8 + i);
            const float x2m = __shfl(p, m, 32) + __shfl(p, m + 16, 32);
            md[i] += x2m;
        }

        // min over columns N: reduce across the 16 lanes of each half-wave
        #pragma unroll
        for (int mask = 8; mask >= 1; mask >>= 1) {
            #pragma unroll
            for (int i = 0; i < 8; ++i)
                md[i] = fmin_raw(md[i], __shfl_xor(md[i], mask, 32));
        }

        if (lane == 0) {
            #pragma unroll
            for (int i = 0; i < 8; ++i) lmin[c][wave][i] = md[i];
        }
        if (lane == 16) {
            #pragma unroll
            for (int i = 0; i < 8; ++i) lmin[c][wave][8 + i] = md[i];
        }
    }
    __syncthreads();

    // cross-wave min + |sqrt - sqrt| + tile sum (threads 0..15 = wave 0 low half)
    if (threadIdx.x < 16) {
        float mp = lmin[0][0][threadIdx.x];
        float mg = lmin[1][0][threadIdx.x];
        #pragma unroll
        for (int w = 1; w < WAVES; ++w) {
            mp = fmin_raw(mp, lmin[0][w][threadIdx.x]);
            mg = fmin_raw(mg, lmin[1][w][threadIdx.x]);
        }
        float diff = fabsf(sqrtf(fmaxf(mp, 0.0f)) - sqrtf(fmaxf(mg, 0.0f)));
        #pragma unroll
        for (int mask = 8; mask >= 1; mask >>= 1)
            diff += __shfl_xor(diff, mask, 32);
        if (threadIdx.x == 0)
            ws[(size_t)b * TILES + blockIdx.x] = diff;
    }
}

// Deterministic final reduction: out[b] = sum(ws[b,:]) / S
__global__ __launch_bounds__(256) void sdf_reduce_kernel(
    const float* __restrict__ ws, float* __restrict__ out)
{
    const int b = blockIdx.x;
    const int t = threadIdx.x;
    __shared__ float sm[256];
    sm[t] = (t < TILES) ? as_global(ws)[(size_t)b * TILES + t] : 0.0f;
    __syncthreads();
    #pragma unroll
    for (int s = 128; s > 0; s >>= 1) {
        if (t < s) sm[t] += sm[t + s];
        __syncthreads();
    }
    if (t == 0) out[b] = sm[0] * (1.0f / (float)S_);
}

extern "C" void kernel_launch(void* const* d_in, const int* in_sizes, int n_in,
                              void* d_out, int out_size, void* d_ws, size_t ws_size,
                              hipStream_t stream) {
    const float* grid_points = (const float*)d_in[0];  // [B,S,3]
    const float* gts         = (const float*)d_in[1];  // [B,N,3]
    const float* preds       = (const float*)d_in[2];  // [B,N,3]
    float* out = (float*)d_out;                        // [B]
    float* ws  = (float*)d_ws;                         // [B,TILES] floats (4 KB)

    dim3 grid(TILES, B_);
    sdf_min_kernel<<<grid, TPB, 0, stream>>>(grid_points, gts, preds, ws);
    sdf_reduce_kernel<<<B_, 256, 0, stream>>>(ws, out);
}